// NLSearch_51694226374947
// MI455X (gfx1250) — compile-verified
//
#include <hip/hip_runtime.h>
#include <math.h>

typedef float v2f __attribute__((ext_vector_type(2)));
typedef float v8f __attribute__((ext_vector_type(8)));

namespace {
constexpr int T_   = 4;
constexpr int C_   = 64;
constexpr int H_   = 192;
constexpr int W_   = 192;
constexpr int CPH  = 32;   // channels per head
constexpr int KTOP = 7;
constexpr int NHh  = 48;
constexpr int NWw  = 48;
constexpr int L_   = 192;  // 3 * 8 * 8 offsets
constexpr int NKK  = 224;  // GEMM K: 7 dy rows * 32 channels
constexpr int AXN  = 80;   // padded A x-extent (need 67)
constexpr int BXN  = 96;   // B x'-extent
constexpr int NR1  = 14;   // vid1 row span: di in [-3,4] + dy in [-3,3]
constexpr int QT   = 16;   // queries per workgroup

constexpr int A_FLOATS = AXN * NKK;        // 17920
constexpr int B_FLOATS = NR1 * CPH * BXN;  // 43008
constexpr int E_FLOATS = 8 * 16 * 32;      // 4096 (per-wave D spill slots)
constexpr int D_FLOATS = QT * L_;          // 3072
constexpr int SMEM_FLOATS = A_FLOATS + B_FLOATS + E_FLOATS + D_FLOATS;
constexpr int QTOT = T_ * NHh * NWw;       // 9216
}

__device__ __forceinline__ int reflect_i(int idx, int n) {
  idx = (idx < 0) ? -idx : idx;
  idx = (idx >= n) ? (2 * (n - 1) - idx) : idx;
  return idx;
}
__device__ __forceinline__ int clamp_i(int v, int lo, int hi) {
  return v < lo ? lo : (v > hi ? hi : v);
}

extern "C" __global__ __launch_bounds__(256)
void nls_wmma_kernel(const float* __restrict__ vid0,
                     const float* __restrict__ vid1,
                     float* __restrict__ vals_out,
                     int* __restrict__ inds_out)
{
  extern __shared__ float smem[];
  float* A_lds = smem;                 // [AXN][NKK]
  float* B_lds = A_lds + A_FLOATS;     // [NR1][16 cpair][BXN][2]
  float* E_lds = B_lds + B_FLOATS;     // [8 waves][16][32]
  float* D_lds = E_lds + E_FLOATS;     // [QT][L_]

  const int bx   = blockIdx.x;
  const int head = bx & 1;
  const int qt   = (bx >> 1) % 3;
  const int qhi  = (bx / 6) % NHh;
  const int t    = bx / (6 * NHh);
  const int qh   = qhi * 4;
  const int xs   = qt * 64 - 3;        // leftmost unreflected x (query col - 3)

  const int tid  = threadIdx.x;
  const int lane = tid & 31;
  const int wave = tid >> 5;
  const int m16  = lane & 15;
  const int hk2  = (lane >> 4) << 1;   // K sub-offset for this half-wave: 0 or 2

  // ---- stage A: vid0 patch rows for 16 queries, K = (dy, c) ----
  for (int i = tid; i < A_FLOATS; i += 256) {
    int xl = i % AXN, kk = i / AXN;
    int dyi = kk >> 5, c = kk & 31;
    int row = reflect_i(qh + dyi - 3, H_);
    int px  = reflect_i(xs + xl, W_);
    A_lds[xl * NKK + kk] =
        vid0[(((size_t)t * C_ + head * CPH + c) * H_ + row) * W_ + px];
  }
  for (int i = tid; i < D_FLOATS; i += 256) D_lds[i] = 0.0f;
  __syncthreads();

  for (int dti = 0; dti < 3; ++dti) {
    const int t1 = clamp_i(t + dti - 1, 0, T_ - 1);
    // ---- stage B: vid1 rows qh-6..qh+7, channel pairs interleaved ----
    for (int i = tid; i < B_FLOATS; i += 256) {
      int xpl  = i % BXN;
      int rest = i / BXN;
      int c = rest & 31, yy = rest >> 5;
      int row = reflect_i(qh + yy - 6, H_);
      int px  = reflect_i(xs - 3 + xpl, W_);
      B_lds[(((yy * 16 + (c >> 1)) * BXN) + xpl) * 2 + (c & 1)] =
          vid1[(((size_t)t1 * C_ + head * CPH + c) * H_ + row) * W_ + px];
    }
    __syncthreads();

    // ---- units: (di, x-tile) -> two 16x16xK GEMMs via chained f32 WMMA ----
    for (int u = wave; u < 40; u += 8) {
      const int dii = u / 5;   // di index 0..7  (di = dii-3)
      const int a   = u % 5;   // x tile
      v8f acc0 = {};
      v8f acc1 = {};
      const float* Ab = &A_lds[(a * 16 + m16) * NKK + hk2];
      for (int s = 0; s < 56; ++s) {
        int kk  = 4 * s + hk2;
        int dyi = kk >> 5;
        int cp  = (kk & 31) >> 1;
        int yy  = dii + dyi;                         // vid1 staged row
        v2f av = *(const v2f*)(Ab + 4 * s);          // A[M, kk], A[M, kk+1]
        const float* Bb = &B_lds[(((yy * 16 + cp) * BXN) + a * 16 + m16) * 2];
        v2f b0 = *(const v2f*)(Bb);                  // B[kk, n], B[kk+1, n]
        v2f b1 = *(const v2f*)(Bb + 32);             // next B tile (+16 cols)
        acc0 = __builtin_amdgcn_wmma_f32_16x16x4_f32(false, av, false, b0,
                                                     (short)0, acc0, false, false);
        acc1 = __builtin_amdgcn_wmma_f32_16x16x4_f32(false, av, false, b1,
                                                     (short)0, acc1, false, false);
      }
      // spill D fragments (VGPR v: M = v + 8*(lane>=16), N = lane&15)
      const int eb  = wave * 512;
      const int rlo = (lane >> 4) << 3;
      for (int v = 0; v < 8; ++v) {
        E_lds[eb + (rlo + v) * 32 + m16]      = acc0[v];
        E_lds[eb + (rlo + v) * 32 + 16 + m16] = acc1[v];
      }
      // banded reduction: dist[jq, dt, di, dj] += E[x, x+dj]
      for (int cell = lane; cell < 128; cell += 32) {
        int mrow = cell >> 3;
        int dji  = cell & 7;                        // dj = dji - 3
        float e  = E_lds[eb + mrow * 32 + (mrow + dji)];
        int xl   = a * 16 + mrow;                   // x = xs + xl
        int r    = (xl + 1) & 3;                    // dx candidates: r, r-4
        int lidx = dti * 64 + dii * 8 + dji;
        int n1 = xl - r - 3;                        // dx = r   -> jq = n1/4
        if (n1 >= 0) { int jq = n1 >> 2; if (jq < QT) atomicAdd(&D_lds[jq * L_ + lidx], e); }
        if (r >= 1) {
          int n2 = xl - r + 1;                      // dx = r-4 -> jq = n2/4
          if (n2 >= 0) { int jq = n2 >> 2; if (jq < QT) atomicAdd(&D_lds[jq * L_ + lidx], e); }
        }
      }
    }
    __syncthreads();
  }

  // ---- fused stable top-K (JAX top_k tie order: first index wins) ----
  if (tid < QT) {
    const int jq = tid;
    float* dq = &D_lds[jq * L_];
    const int q     = (t * NHh + qhi) * NWw + (qt * QT + jq);
    const int obase = (head * QTOT + q) * KTOP;
    const int qw    = qt * 64 + jq * 4;
    for (int k = 0; k < KTOP; ++k) {
      float bv = -INFINITY; int bl = 0;
      for (int l = 0; l < L_; ++l) {
        float v = dq[l];
        if (v > bv) { bv = v; bl = l; }
      }
      dq[bl] = -INFINITY;
      vals_out[obase + k] = bv;
      int dtv = (bl >> 6) - 1;
      int div = ((bl >> 3) & 7) - 3;
      int djv = (bl & 7) - 3;
      inds_out[(obase + k) * 3 + 0] = clamp_i(t + dtv, 0, T_ - 1);
      inds_out[(obase + k) * 3 + 1] = reflect_i(qh + div, H_);
      inds_out[(obase + k) * 3 + 2] = reflect_i(qw + djv, W_);
    }
  }
}

extern "C" void kernel_launch(void* const* d_in, const int* in_sizes, int n_in,
                              void* d_out, int out_size, void* d_ws, size_t ws_size,
                              hipStream_t stream) {
  (void)in_sizes; (void)n_in; (void)d_ws; (void)ws_size; (void)out_size;
  const float* vid0 = (const float*)d_in[0];
  const float* vid1 = (const float*)d_in[1];
  float* vals = (float*)d_out;
  int*   inds = (int*)d_out + 2 * QTOT * KTOP;   // vals (f32) then inds (i32)

  const size_t shmem = (size_t)SMEM_FLOATS * sizeof(float);  // ~266 KB < 320 KB/WGP
  hipFuncSetAttribute((const void*)nls_wmma_kernel,
                      hipFuncAttributeMaxDynamicSharedMemorySize, (int)shmem);

  const int nblocks = T_ * NHh * 3 * 2;  // 1152 WGs: (t, qh, qtile, head)
  nls_wmma_kernel<<<dim3(nblocks), dim3(256), shmem, stream>>>(vid0, vid1, vals, inds);
}